// GCN_6588479832097
// MI455X (gfx1250) — compile-verified
//
#include <hip/hip_runtime.h>
#include <hip/hip_bf16.h>
#include <math.h>

// ---------------------------------------------------------------------------
// GCN forward on MI455X (gfx1250, wave32).
// GEMMs via V_WMMA_F32_16X16X4_F32 (exact fp32 matrix path), 2 row tiles per
// wave (B fragments loaded once, reused), self-loop scaling fused into the
// GEMM epilogue. Message passing via prefetched float4 gathers +
// global_atomic_add_f32 scatters (feature maps are L2-resident: <=45MB vs
// 192MB L2).
// ---------------------------------------------------------------------------

typedef __attribute__((ext_vector_type(2))) float v2f;
typedef __attribute__((ext_vector_type(8))) float v8f;

#define GCN_NODES   100000
#define GCN_NIN     128
#define GCN_GRAPHS  64

// ---------------- utility fills ----------------
__global__ void gcn_fill_f32(float* __restrict__ p, float v, int n) {
    int i = blockIdx.x * blockDim.x + threadIdx.x;
    if (i < n) p[i] = v;
}

// ---------------- degree / dinv ----------------
__global__ void gcn_deg_edges(const int* __restrict__ dst, float* __restrict__ deg, int E) {
    int e = blockIdx.x * blockDim.x + threadIdx.x;
    if (e < E) atomicAdd(&deg[dst[e]], 1.0f);
}

__global__ void gcn_deg_to_dinv(float* __restrict__ deg, int n) {
    int i = blockIdx.x * blockDim.x + threadIdx.x;
    if (i < n) {
        float d = deg[i];            // >= 1 (self loop)
        deg[i] = rsqrtf(d);
    }
}

// ---------------- weight padding: W[K,C] -> Wp[Kp,Cp] (zero pad) ----------------
__global__ void gcn_pad_w(const float* __restrict__ W, float* __restrict__ Wp,
                          int K, int C, int Kp, int Cp) {
    int i = blockIdx.x * blockDim.x + threadIdx.x;
    if (i >= Kp * Cp) return;
    int kp = i / Cp, cp = i % Cp;
    Wp[i] = (kp < K && cp < C) ? W[kp * C + cp] : 0.0f;
}

// ---------------- WMMA fp32 GEMM + fused self-loop scale ----------------
// H[N,ldc] = A[N,lda] x B[K,ldb];  Agg[N,ldc] = H * dinv[row]^2
// grid.x = N/32 (2 row tiles per wave); blockDim.x = 32 * (ldc/16);
// wave w -> column tile w. B fragments are loaded once per k-step and feed
// two v_wmma issues (two row tiles).
__global__ void gcn_gemm_wmma_f32(const float* __restrict__ A, const float* __restrict__ B,
                                  float* __restrict__ H, float* __restrict__ Agg,
                                  const float* __restrict__ dinv,
                                  int K, int lda, int ldb, int ldc) {
    const int lane = threadIdx.x & 31;
    const int colTile = threadIdx.x >> 5;
    const int row0 = blockIdx.x << 5;               // 2 row tiles of 16
    const int col0 = colTile << 4;
    const int m  = lane & 15;
    const int hi = lane >> 4;                       // 0: K pair {0,1}; 1: K pair {2,3}

    v8f acc0 = {};
    v8f acc1 = {};
    const float* __restrict__ arow0 = A + (size_t)(row0 + m) * lda;
    const float* __restrict__ arow1 = A + (size_t)(row0 + 16 + m) * lda;

    for (int k = 0; k < K; k += 4) {
        // B fragment (4x16 f32): row-striped across lanes, half-lane K split.
        const int kb = k + 2 * hi;
        v2f b;
        b.x = B[(size_t)kb       * ldb + col0 + m];
        b.y = B[(size_t)(kb + 1) * ldb + col0 + m];
        // A fragments (16x4 f32): VGPR0 = X[m][k+2*hi], VGPR1 = X[m][k+2*hi+1]
        const float2 a0v = *reinterpret_cast<const float2*>(arow0 + k + 2 * hi);
        const float2 a1v = *reinterpret_cast<const float2*>(arow1 + k + 2 * hi);
        v2f a0; a0.x = a0v.x; a0.y = a0v.y;
        v2f a1; a1.x = a1v.x; a1.y = a1v.y;
        acc0 = __builtin_amdgcn_wmma_f32_16x16x4_f32(false, a0, false, b,
                                                     (short)0, acc0, false, false);
        acc1 = __builtin_amdgcn_wmma_f32_16x16x4_f32(false, a1, false, b,
                                                     (short)0, acc1, false, false);
    }

    // C/D layout: VGPR v -> row (v + 8*hi), col = lane&15.
    // Fused epilogue: write H and the self-loop-initialized aggregation buffer.
    #pragma unroll
    for (int v = 0; v < 8; ++v) {
        int r0 = row0 + v + 8 * hi;
        int r1 = r0 + 16;
        float d0 = dinv[r0];
        float d1 = dinv[r1];
        size_t o0 = (size_t)r0 * ldc + col0 + m;
        size_t o1 = (size_t)r1 * ldc + col0 + m;
        H[o0]   = acc0[v];
        H[o1]   = acc1[v];
        Agg[o0] = acc0[v] * (d0 * d0);
        Agg[o1] = acc1[v] * (d1 * d1);
    }
}

// ---------------- aggregation: edge scatter out[dst] += h[src]*norm ----------------
__global__ void gcn_edge_scatter(const int* __restrict__ src, const int* __restrict__ dst,
                                 const float* __restrict__ dinv, const float* __restrict__ h,
                                 float* __restrict__ out, int ldc, int cw, int E) {
    int e = blockIdx.x * blockDim.x + threadIdx.x;
    if (e >= E) return;
    int s = src[e], d = dst[e];
    const float* __restrict__ hs = h + (size_t)s * ldc;
    // Warm the gathered row (global_prefetch_b8) while dinv loads resolve.
    __builtin_prefetch(hs, 0, 0);
    float norm = dinv[s] * dinv[d];
    float* __restrict__ od = out + (size_t)d * ldc;
    for (int c = 0; c < cw; c += 4) {
        float4 v = *reinterpret_cast<const float4*>(hs + c);
        atomicAdd(&od[c + 0], v.x * norm);
        atomicAdd(&od[c + 1], v.y * norm);
        atomicAdd(&od[c + 2], v.z * norm);
        atomicAdd(&od[c + 3], v.w * norm);
    }
}

// ---------------- bias + relu (in place) ----------------
__global__ void gcn_bias_relu(float* __restrict__ x, const float* __restrict__ b,
                              int n, int ldc, int creal) {
    int i = blockIdx.x * blockDim.x + threadIdx.x;
    if (i >= n * ldc) return;
    int c = i % ldc;
    float bias = (c < creal) ? b[c] : 0.0f;
    float v = x[i] + bias;
    x[i] = fmaxf(v, 0.0f);
}

// ---------------- pooling: counts + segment sums ----------------
__global__ void gcn_pool_cnt(const int* __restrict__ batch, float* __restrict__ cnt, int n) {
    int i = blockIdx.x * blockDim.x + threadIdx.x;
    if (i < n) atomicAdd(&cnt[batch[i]], 1.0f);
}

// one thread per (node, 4-col group): 100 cols -> 25 groups
__global__ void gcn_pool_sum(const int* __restrict__ batch, const float* __restrict__ h,
                             float* __restrict__ sums, int n, int ldc, int creal4) {
    int i = blockIdx.x * blockDim.x + threadIdx.x;
    int ngrp = creal4 >> 2;
    if (i >= n * ngrp) return;
    int node = i / ngrp;
    int c = (i % ngrp) << 2;
    int g = batch[node];
    float4 v = *reinterpret_cast<const float4*>(h + (size_t)node * ldc + c);
    float* sg = sums + (size_t)g * creal4 + c;
    atomicAdd(&sg[0], v.x);
    atomicAdd(&sg[1], v.y);
    atomicAdd(&sg[2], v.z);
    atomicAdd(&sg[3], v.w);
}

// ---------------- head: pooled = sums/cnt; out = (pooled@Wf1+bf1)@Wf2+bf2 ----------------
// single block, 256 threads; staging in LDS
__global__ void gcn_head(const float* __restrict__ sums, const float* __restrict__ cnt,
                         const float* __restrict__ Wf1, const float* __restrict__ bf1,
                         const float* __restrict__ Wf2, const float* __restrict__ bf2,
                         float* __restrict__ out) {
    __shared__ float pooled[GCN_GRAPHS * 100];
    __shared__ float t1[GCN_GRAPHS * 50];
    int tid = threadIdx.x;

    for (int i = tid; i < GCN_GRAPHS * 100; i += blockDim.x) {
        int g = i / 100;
        float c = fmaxf(cnt[g], 1.0f);
        pooled[i] = sums[i] / c;
    }
    __syncthreads();

    for (int i = tid; i < GCN_GRAPHS * 50; i += blockDim.x) {
        int g = i / 50, j = i % 50;
        float s = bf1[j];
        const float* pg = pooled + g * 100;
        for (int k = 0; k < 100; ++k) s += pg[k] * Wf1[k * 50 + j];
        t1[i] = s;
    }
    __syncthreads();

    for (int i = tid; i < GCN_GRAPHS * 2; i += blockDim.x) {
        int g = i / 2, o = i % 2;
        float s = bf2[o];
        const float* tg = t1 + g * 50;
        for (int j = 0; j < 50; ++j) s += tg[j] * Wf2[j * 2 + o];
        out[i] = s;
    }
}

// ---------------------------------------------------------------------------
extern "C" void kernel_launch(void* const* d_in, const int* in_sizes, int n_in,
                              void* d_out, int out_size, void* d_ws, size_t ws_size,
                              hipStream_t stream) {
    const float* x     = (const float*)d_in[0];
    const int*   eidx  = (const int*)d_in[1];
    const int*   batch = (const int*)d_in[2];
    const float* W1 = (const float*)d_in[3];  const float* b1 = (const float*)d_in[4];
    const float* W2 = (const float*)d_in[5];  const float* b2 = (const float*)d_in[6];
    const float* W3 = (const float*)d_in[7];  const float* b3 = (const float*)d_in[8];
    const float* W4 = (const float*)d_in[9];  const float* b4 = (const float*)d_in[10];
    const float* Wf1 = (const float*)d_in[11]; const float* bf1 = (const float*)d_in[12];
    const float* Wf2 = (const float*)d_in[13]; const float* bf2 = (const float*)d_in[14];

    const int N = in_sizes[0] / GCN_NIN;       // 100000
    const int E = in_sizes[1] / 2;             // 1600000
    const int* src = eidx;
    const int* dst = eidx + E;

    // ---- workspace carve-up (floats) ----
    float* ws = (float*)d_ws;
    float* B0   = ws;                 ws += (size_t)N * 112;   // GEMM output H
    float* B1   = ws;                 ws += (size_t)N * 112;   // aggregation / activation
    float* dinv = ws;                 ws += N;                 // deg -> dinv (in place)
    float* W1p  = ws;                 ws += 128 * 64;
    float* W2p  = ws;                 ws += 64 * 64;
    float* W3p  = ws;                 ws += 64 * 64;
    float* W4p  = ws;                 ws += 64 * 112;
    float* sums = ws;                 ws += GCN_GRAPHS * 100;
    float* cnt  = ws;                 ws += GCN_GRAPHS;

    const int TB = 256;
    auto blks = [](long long n, int tb) { return (int)((n + tb - 1) / tb); };

    // ---- degrees (self loop = 1) and D^-1/2 ----
    gcn_fill_f32<<<blks(N, TB), TB, 0, stream>>>(dinv, 1.0f, N);
    gcn_deg_edges<<<blks(E, TB), TB, 0, stream>>>(dst, dinv, E);
    gcn_deg_to_dinv<<<blks(N, TB), TB, 0, stream>>>(dinv, N);

    // ---- pad weights into WMMA-friendly shapes ----
    gcn_pad_w<<<blks(128 * 64, TB), TB, 0, stream>>>(W1, W1p, 128, 50, 128, 64);
    gcn_pad_w<<<blks(64 * 64, TB), TB, 0, stream>>>(W2, W2p, 50, 50, 64, 64);
    gcn_pad_w<<<blks(64 * 64, TB), TB, 0, stream>>>(W3, W3p, 50, 50, 64, 64);
    gcn_pad_w<<<blks(64 * 112, TB), TB, 0, stream>>>(W4, W4p, 50, 100, 64, 112);

    const int rowBlocks = N / 32;   // 3125 exactly (2 row tiles per wave)

    // ---- layer descriptor loop ----
    struct Layer { const float* in; int K; int lda; const float* Wp; int cpad; int creal; const float* b; };
    Layer layers[4] = {
        { x,  128, 128, W1p, 64,  50,  b1 },
        { B1, 64,  64,  W2p, 64,  50,  b2 },
        { B1, 64,  64,  W3p, 64,  50,  b3 },
        { B1, 64,  64,  W4p, 112, 100, b4 },
    };

    for (int l = 0; l < 4; ++l) {
        const Layer& L = layers[l];
        int nct = L.cpad / 16;
        // H = act @ Wp (WMMA fp32), fused: B1 = H * dinv^2 (self-loop init)
        gcn_gemm_wmma_f32<<<rowBlocks, 32 * nct, 0, stream>>>(L.in, L.Wp, B0, B1, dinv,
                                                              L.K, L.lda, L.cpad, L.cpad);
        // edge scatter: B1[dst] += H[src] * dinv[src]*dinv[dst]
        int cw = (L.creal + 3) & ~3;
        gcn_edge_scatter<<<blks(E, TB), TB, 0, stream>>>(src, dst, dinv, B0, B1, L.cpad, cw, E);
        // bias + relu (pad cols stay 0)
        gcn_bias_relu<<<blks((long long)N * L.cpad, TB), TB, 0, stream>>>(B1, L.b, N, L.cpad, L.creal);
    }

    // ---- global mean pool over graphs ----
    gcn_fill_f32<<<blks(GCN_GRAPHS * 100 + GCN_GRAPHS, TB), TB, 0, stream>>>(sums, 0.0f, GCN_GRAPHS * 100 + GCN_GRAPHS);
    gcn_pool_cnt<<<blks(N, TB), TB, 0, stream>>>(batch, cnt, N);
    gcn_pool_sum<<<blks((long long)N * 25, TB), TB, 0, stream>>>(batch, B1, sums, N, 112, 100);

    // ---- dense head ----
    gcn_head<<<1, 256, 0, stream>>>(sums, cnt, Wf1, bf1, Wf2, bf2, (float*)d_out);
}